// SqueezeformerCTC_90640989815503
// MI455X (gfx1250) — compile-verified
//
#include <hip/hip_runtime.h>
#include <hip/hip_bf16.h>
#include <stdint.h>

// Problem constants (from reference setup_inputs)
#define B_DIM 32
#define T_DIM 4096
#define D_DIM 256
#define THRESH 0.5f
#define MIN_KEEP 4

#define PLAN_THREADS 1024
#define CHUNK (T_DIM / PLAN_THREADS)   // 4 contiguous time steps per thread

#define TILE_ROWS 16                   // rows per TDM tile (gather-mode index limit)

typedef float f4  __attribute__((ext_vector_type(4)));
typedef int   v4i __attribute__((ext_vector_type(4)));
typedef int   v8i __attribute__((ext_vector_type(8)));

#define AS3 __attribute__((address_space(3)))

__device__ __forceinline__ void wait_tensor0() {
#if __has_builtin(__builtin_amdgcn_s_wait_tensorcnt)
  __builtin_amdgcn_s_wait_tensorcnt(0);
#else
  asm volatile("s_wait_tensorcnt 0" ::: "memory");
#endif
}

// TDM issue helpers: descriptor groups live in SGPRs ("s" constraints force
// SGPR tuples: v4i -> s[n:n+3], v8i -> s[n:n+7]).
__device__ __forceinline__ void tdm_load(v4i g0, v8i g1, v4i g2, v4i g3) {
  asm volatile("tensor_load_to_lds %0, %1, %2, %3"
               :: "s"(g0), "s"(g1), "s"(g2), "s"(g3) : "memory");
}
__device__ __forceinline__ void tdm_store(v4i g0, v8i g1, v4i g2, v4i g3) {
  asm volatile("tensor_store_from_lds %0, %1, %2, %3"
               :: "s"(g0), "s"(g1), "s"(g2), "s"(g3) : "memory");
}

// Block-wide exclusive scan (Hillis-Steele) over PLAN_THREADS ints.
__device__ __forceinline__ int block_exscan(int val, int* sm, int* total_out) {
  const int tid = threadIdx.x;
  __syncthreads();               // protect sm from previous use
  sm[tid] = val;
  __syncthreads();
  for (int off = 1; off < PLAN_THREADS; off <<= 1) {
    int other = (tid >= off) ? sm[tid - off] : 0;
    __syncthreads();
    sm[tid] += other;
    __syncthreads();
  }
  int incl = sm[tid];
  *total_out = sm[PLAN_THREADS - 1];
  return incl - val;
}

// One block per batch row: reproduce reference keep-mask logic exactly and
// emit compacted gather indices (u16, strictly increasing) + pruned length.
__global__ __launch_bounds__(PLAN_THREADS)
void prune_plan_kernel(const float* __restrict__ bp,
                       const int* __restrict__ lengths,
                       unsigned short* __restrict__ gidx,   // [B, T] (first len used)
                       int* __restrict__ wlen,              // [B] for copy kernel
                       int* __restrict__ out_len_tail)      // [B] tail of d_out (int bits)
{
  __shared__ int s_scan[PLAN_THREADS];
  __shared__ unsigned long long s_red[PLAN_THREADS];

  const int b   = blockIdx.x;
  const int tid = threadIdx.x;
  const int L   = lengths[b];

  int required = L;                       // clip(L, 0, MIN_KEEP)
  if (required < 0) required = 0;
  if (required > MIN_KEEP) required = MIN_KEEP;

  const int t0 = tid * CHUNK;
  f4 vv = ((const f4*)(bp + (size_t)b * T_DIM))[tid];
  float v[CHUNK] = { vv.x, vv.y, vv.z, vv.w };

  bool keep0[CHUNK], cand[CHUNK];
  int keepCnt = 0, candCnt = 0;
#pragma unroll
  for (int k = 0; k < CHUNK; ++k) {
    bool valid = (t0 + k) < L;
    bool kp    = valid && (v[k] < THRESH);
    keep0[k] = kp;
    cand[k]  = valid && !kp;              // valid-but-not-kept candidates
    keepCnt += kp ? 1 : 0;
    candCnt += cand[k] ? 1 : 0;
  }

  int keepTotal;
  (void)block_exscan(keepCnt, s_scan, &keepTotal);   // uniform total

  bool fk[CHUNK];
  if (keepTotal >= MIN_KEEP) {
#pragma unroll
    for (int k = 0; k < CHUNK; ++k) fk[k] = keep0[k];
  } else if (keepTotal > 0) {
    // pad with earliest valid-but-not-kept frames
    int additional = required - keepTotal;
    if (additional < 0) additional = 0;
    int candTotal;
    int cp = block_exscan(candCnt, s_scan, &candTotal);  // exclusive candidate rank
#pragma unroll
    for (int k = 0; k < CHUNK; ++k) {
      fk[k] = keep0[k] || (cand[k] && (cp < additional));
      cp += cand[k] ? 1 : 0;
    }
  } else {
    // zero kept: take `required` frames with smallest blank prob
    // (value-then-index order matches jax.lax.top_k tie-breaking; bp >= 0 so
    //  float bit pattern order == numeric order)
    float mv[CHUNK];
#pragma unroll
    for (int k = 0; k < CHUNK; ++k) {
      mv[k] = ((t0 + k) < L) ? v[k] : __builtin_inff();
      fk[k] = false;
    }
    for (int j = 0; j < MIN_KEEP; ++j) {
      unsigned long long best = ~0ull;
#pragma unroll
      for (int k = 0; k < CHUNK; ++k) {
        unsigned long long key =
            ((unsigned long long)__float_as_uint(mv[k]) << 32) | (unsigned)(t0 + k);
        if (key < best) best = key;
      }
      __syncthreads();
      s_red[tid] = best;
      __syncthreads();
      for (int off = PLAN_THREADS / 2; off > 0; off >>= 1) {
        if (tid < off && s_red[tid + off] < s_red[tid]) s_red[tid] = s_red[tid + off];
        __syncthreads();
      }
      int widx = (int)(s_red[0] & 0xFFFFFFFFu);    // uniform winner index
      if (widx >= t0 && widx < t0 + CHUNK) {
        int k = widx - t0;
        mv[k] = __builtin_inff();
        if (j < required) fk[k] = true;
      }
    }
  }

  // stable compaction positions for final keep mask
  int finalCnt = 0;
#pragma unroll
  for (int k = 0; k < CHUNK; ++k) finalCnt += fk[k] ? 1 : 0;
  int finTot;
  int pos = block_exscan(finalCnt, s_scan, &finTot);

  unsigned short* grow = gidx + (size_t)b * T_DIM;
#pragma unroll
  for (int k = 0; k < CHUNK; ++k) {
    if (fk[k]) { grow[pos] = (unsigned short)(t0 + k); ++pos; }
  }

  if (tid == 0) {
    int finalLen = (keepTotal < MIN_KEEP) ? required : keepTotal;
    wlen[b] = finalLen;
    out_len_tail[b] = finalLen;
  }
}

// TDM gather kernel: one wave per (batch, 16-output-row tile).
//  - TENSOR_LOAD_TO_LDS in gather mode pulls the nvalid indexed 1KB rows of x
//    into a compact 16-row LDS tile (indices strictly increasing).
//  - TENSOR_STORE_FROM_LDS writes the tile contiguously to the output.
//  - rows >= pruned_len are zero-filled with NT b128 stores.
__global__ __launch_bounds__(32)
void gather_rows_tdm_kernel(const float* __restrict__ x,
                            const unsigned short* __restrict__ gidx,
                            const int* __restrict__ wlen,
                            float* __restrict__ out)
{
  __shared__ __align__(128) float lds[TILE_ROWS * D_DIM];   // 16 KB tile

  const int b    = blockIdx.y;
  const int j0   = blockIdx.x * TILE_ROWS;
  const int lane = threadIdx.x;                              // 0..31
  const int len  = wlen[b];

  int nvalid = len - j0;
  if (nvalid < 0) nvalid = 0;
  if (nvalid > TILE_ROWS) nvalid = TILE_ROWS;

  if (nvalid > 0) {
    const unsigned lds_off = (unsigned)(size_t)(AS3 char*)lds;   // bytes, wg-relative
    const unsigned long long xaddr =
        (unsigned long long)(size_t)(x + (size_t)b * T_DIM * D_DIM);
    const unsigned long long oaddr =
        (unsigned long long)(size_t)(out + ((size_t)b * T_DIM + j0) * D_DIM);

    // 16 gather row indices (u16, 32B aligned) -> SGPR groups 2/3.
    // Layout matches D#: row_index_{2k} in dword[k][15:0], {2k+1} in [31:16].
    union { v8i v; struct { v4i lo, hi; } s; } idx;
    idx.v = *(const v8i*)(gidx + (size_t)b * T_DIM + j0);

    // ---- D# group 1 (shared by load and store) ----
    // [17:16] data_size=2 (4B) | tensor_dim0=256 @ bits79:48 |
    // tensor_dim1=4096 @ 111:80 | tile_dim0=256 @ 127:112 |
    // tile_dim1=nvalid @ 143:128 | tensor_dim0_stride=256 @ 207:160
    v8i g1;
    g1[0] = 0x00020000;                                  // data_size = 4 bytes
    g1[1] = (D_DIM & 0xFFFF) << 16;                      // tensor_dim0 lo16
    g1[2] = (D_DIM >> 16) | ((T_DIM & 0xFFFF) << 16);    // dim0 hi16 | dim1 lo16
    g1[3] = (T_DIM >> 16) | (D_DIM << 16);               // dim1 hi16 | tile_dim0
    g1[4] = nvalid;                                      // tile_dim1 (tile_dim2=0)
    g1[5] = D_DIM;                                       // tensor_dim0_stride lo32
    g1[6] = 0;                                           // stride0 hi | stride1 lo
    g1[7] = 0;                                           // stride1 hi

    // ---- D# group 0: load (gather mode) ----
    v4i g0l, g0s;
    g0l[0] = (int)0x80000001u;                           // count=1 | gather_mode
    g0l[1] = (int)lds_off;
    g0l[2] = (int)(xaddr & 0xFFFFFFFFu);
    g0l[3] = (int)(((xaddr >> 32) & 0x01FFFFFFu) | 0x80000000u);  // type=2
    // ---- D# group 0: store (normal 2D tile) ----
    g0s[0] = 1;                                          // count=1
    g0s[1] = (int)lds_off;
    g0s[2] = (int)(oaddr & 0xFFFFFFFFu);
    g0s[3] = (int)(((oaddr >> 32) & 0x01FFFFFFu) | 0x80000000u);  // type=2

    v4i zero4 = { 0, 0, 0, 0 };                          // groups 2/3 for 2D store

    tdm_load(g0l, g1, idx.s.lo, idx.s.hi);
    wait_tensor0();                                      // tile resident in LDS
    tdm_store(g0s, g1, zero4, zero4);
    wait_tensor0();                                      // LDS reusable / done
  }

  // zero-fill rows [nvalid, TILE_ROWS) of this tile
  const int zrows = TILE_ROWS - nvalid;
  if (zrows > 0) {
    float* zbase = out + ((size_t)b * T_DIM + j0 + nvalid) * D_DIM;
    const int total_f4 = zrows * (D_DIM / 4);
    f4 z = { 0.f, 0.f, 0.f, 0.f };
    for (int i = lane; i < total_f4; i += 32)
      __builtin_nontemporal_store(z, (f4*)zbase + i);
  }
}

extern "C" void kernel_launch(void* const* d_in, const int* in_sizes, int n_in,
                              void* d_out, int out_size, void* d_ws, size_t ws_size,
                              hipStream_t stream) {
  (void)in_sizes; (void)n_in; (void)out_size; (void)ws_size;

  const float* x       = (const float*)d_in[0];          // [B,T,D] f32
  const int*   lengths = (const int*)d_in[1];            // [B] i32
  const float* bp      = (const float*)d_in[2];          // [B,T] f32
  float*       out     = (float*)d_out;                  // [B,T,D] f32 ++ [B] i32 (bits)

  unsigned short* gidx = (unsigned short*)d_ws;                                  // 256 KB
  int* wlen = (int*)((char*)d_ws + (size_t)B_DIM * T_DIM * sizeof(unsigned short));
  int* out_len_tail = (int*)(out + (size_t)B_DIM * T_DIM * D_DIM);

  prune_plan_kernel<<<B_DIM, PLAN_THREADS, 0, stream>>>(bp, lengths, gidx, wlen, out_len_tail);

  dim3 grid(T_DIM / TILE_ROWS, B_DIM);                   // (256, 32) waves
  gather_rows_tdm_kernel<<<grid, 32, 0, stream>>>(x, gidx, wlen, out);
}